// SelectiveSSM_48954037240249
// MI455X (gfx1250) — compile-verified
//
#include <hip/hip_runtime.h>

// Selective SSM (S4D) fused chunked-scan implementation for gfx1250 (MI455X).
// L=2048, D=1024, N=16. Memory-bound: hs output alone is 128 MB.
//
// Kernel pipeline (all on `stream`, in-order):
//   1) proj_wmma_kernel : proj[l] = xs[l,:]@WD + bD  via V_WMMA_F32_16X16X4_F32
//   2) phaseA_kernel    : per-(chunk,d) local scan -> (A-product, h_end) per state
//   3) phaseB_kernel    : per-(d,n) combine 64 chunk states -> exclusive prefix h_in
//   4) phaseC_kernel    : re-run chunk from h_in, write hs (b128 stores) and ys

#define LSEQ   2048
#define DCH    1024
#define NST    16
#define CHUNK  32
#define NCHUNK (LSEQ / CHUNK)   // 64

typedef float v2f __attribute__((ext_vector_type(2)));
typedef float v8f __attribute__((ext_vector_type(8)));

__device__ __forceinline__ float softplus_f(float z) {
  // numerically stable softplus, matches jax.nn.softplus
  return (z > 0.f) ? (z + log1pf(__expf(-z))) : log1pf(__expf(z));
}

// ---------------------------------------------------------------------------
// Kernel 1: proj[l] = sum_k xs[l,k] * WD[k] + bD
// One wave (32 lanes) per 16 rows of L. A = xs tile (16x4 f32), B = WD
// broadcast to all 16 columns (4x16). Since every B column is identical, the
// D tile is N-invariant; lanes 0 and 16 extract rows 0..7 / 8..15.
// A-matrix 16x4 f32 layout (ISA 7.12.2): lanes 0-15 = M, VGPR0/1 = K0/K1;
// lanes 16-31 = M, VGPR0/1 = K2/K3.  B assumed symmetric (K rows striped the
// same way) -- irrelevant to N since all columns are equal.
// ---------------------------------------------------------------------------
__global__ void __launch_bounds__(32) proj_wmma_kernel(
    const float* __restrict__ xs, const float* __restrict__ WD,
    const float* __restrict__ bD, float* __restrict__ proj) {
  const int row0 = blockIdx.x * 16;
  const int lane = threadIdx.x & 31;
  const int half = lane >> 4;   // 0: K=0,1   1: K=2,3
  const int m    = lane & 15;   // row within tile
  const float* __restrict__ xrow = xs + (size_t)(row0 + m) * DCH;

  v8f acc = {};
  for (int k = 0; k < DCH; k += 4) {
    v2f a, b;
    a.x = xrow[k + 2 * half + 0];
    a.y = xrow[k + 2 * half + 1];
    b.x = WD[k + 2 * half + 0];
    b.y = WD[k + 2 * half + 1];
    // (neg_a, A, neg_b, B, c_mod, C, reuse_a, reuse_b)
    acc = __builtin_amdgcn_wmma_f32_16x16x4_f32(
        false, a, false, b, (short)0, acc, false, false);
  }
  // C/D layout: VGPR j -> M=j (lanes 0-15) or M=j+8 (lanes 16-31); all N equal.
  if (m == 0) {
    const float bd = bD[0];
#pragma unroll
    for (int j = 0; j < 8; ++j)
      proj[row0 + half * 8 + j] = acc[j] + bd;
  }
}

// ---------------------------------------------------------------------------
// Kernel 2: per-(chunk, d) local scan from h=0. Emits per-chunk A-product and
// h_end for each of the N=16 states. 64*1024 = 65536 threads.
// ---------------------------------------------------------------------------
__global__ void __launch_bounds__(256) phaseA_kernel(
    const float* __restrict__ xs, const float* __restrict__ proj,
    const float* __restrict__ WB, const float* __restrict__ bB,
    const float* __restrict__ logA, const float* __restrict__ delta_param,
    float* __restrict__ Aprod, float* __restrict__ Hend) {
  const int d = blockIdx.x * blockDim.x + threadIdx.x;
  const int c = blockIdx.y;

  float Ad[NST], ap[NST], h[NST], wb[NST], bb[NST];
  const float dp = delta_param[d];
#pragma unroll
  for (int n = 0; n < NST; ++n) {
    Ad[n] = -__expf(logA[d * NST + n]);
    wb[n] = WB[n];
    bb[n] = bB[n];
    ap[n] = 1.f;
    h[n]  = 0.f;
  }

  const int l0 = c * CHUNK;
  for (int t = 0; t < CHUNK; ++t) {
    const int l = l0 + t;
    const float x = xs[(size_t)l * DCH + d];
    const float delta = softplus_f(dp + proj[l]);
#pragma unroll
    for (int n = 0; n < NST; ++n) {
      const float lA = delta * Ad[n];           // < 0 always
      const float Ab = __expf(lA);
      const float Bb = (Ab - 1.f) / lA * delta * (x * wb[n] + bb[n]);
      h[n]  = Ab * h[n] + Bb * x;
      ap[n] *= Ab;
    }
  }

  float4* __restrict__ Ap4 = (float4*)(Aprod + ((size_t)c * DCH + d) * NST);
  float4* __restrict__ He4 = (float4*)(Hend  + ((size_t)c * DCH + d) * NST);
#pragma unroll
  for (int q = 0; q < 4; ++q) {
    Ap4[q] = make_float4(ap[4 * q], ap[4 * q + 1], ap[4 * q + 2], ap[4 * q + 3]);
    He4[q] = make_float4(h[4 * q],  h[4 * q + 1],  h[4 * q + 2],  h[4 * q + 3]);
  }
}

// ---------------------------------------------------------------------------
// Kernel 3: per-(d,n) sequential combine over the 64 chunks -> exclusive
// prefix (incoming state h_in for each chunk). 16384 threads, 64 steps.
// ---------------------------------------------------------------------------
__global__ void __launch_bounds__(256) phaseB_kernel(
    const float* __restrict__ Aprod, const float* __restrict__ Hend,
    float* __restrict__ Hin) {
  const int i = blockIdx.x * blockDim.x + threadIdx.x;  // 0 .. D*N-1
  float carry = 0.f;
  for (int c = 0; c < NCHUNK; ++c) {
    const size_t off = (size_t)c * DCH * NST + i;
    Hin[off] = carry;
    carry = Aprod[off] * carry + Hend[off];
  }
}

// ---------------------------------------------------------------------------
// Kernel 4: re-run each chunk from its true incoming state; fuse Cs
// contraction; stream hs with contiguous float4 stores and write ys.
// ---------------------------------------------------------------------------
__global__ void __launch_bounds__(256) phaseC_kernel(
    const float* __restrict__ xs, const float* __restrict__ proj,
    const float* __restrict__ WB, const float* __restrict__ bB,
    const float* __restrict__ WC, const float* __restrict__ bC,
    const float* __restrict__ logA, const float* __restrict__ delta_param,
    const float* __restrict__ Hin,
    float* __restrict__ ys, float* __restrict__ hs) {
  const int d = blockIdx.x * blockDim.x + threadIdx.x;
  const int c = blockIdx.y;

  float Ad[NST], h[NST], wb[NST], bb[NST], wc[NST], bc[NST];
  const float dp = delta_param[d];
#pragma unroll
  for (int n = 0; n < NST; ++n) {
    Ad[n] = -__expf(logA[d * NST + n]);
    wb[n] = WB[n];
    bb[n] = bB[n];
    wc[n] = WC[n];
    bc[n] = bC[n];
  }
  const float4* __restrict__ Hi4 = (const float4*)(Hin + ((size_t)c * DCH + d) * NST);
#pragma unroll
  for (int q = 0; q < 4; ++q) {
    const float4 v = Hi4[q];
    h[4 * q] = v.x; h[4 * q + 1] = v.y; h[4 * q + 2] = v.z; h[4 * q + 3] = v.w;
  }

  const int l0 = c * CHUNK;
  for (int t = 0; t < CHUNK; ++t) {
    const int l = l0 + t;
    const float x = xs[(size_t)l * DCH + d];
    const float delta = softplus_f(dp + proj[l]);
    float y = 0.f;
#pragma unroll
    for (int n = 0; n < NST; ++n) {
      const float lA = delta * Ad[n];
      const float Ab = __expf(lA);
      const float Bb = (Ab - 1.f) / lA * delta * (x * wb[n] + bb[n]);
      h[n] = Ab * h[n] + Bb * x;
      y += h[n] * (x * wc[n] + bc[n]);
    }
    float4* __restrict__ hs4 = (float4*)(hs + ((size_t)l * DCH + d) * NST);
#pragma unroll
    for (int q = 0; q < 4; ++q)
      hs4[q] = make_float4(h[4 * q], h[4 * q + 1], h[4 * q + 2], h[4 * q + 3]);
    ys[(size_t)l * DCH + d] = y;
  }
}

// ---------------------------------------------------------------------------
extern "C" void kernel_launch(void* const* d_in, const int* in_sizes, int n_in,
                              void* d_out, int out_size, void* d_ws, size_t ws_size,
                              hipStream_t stream) {
  const float* xs     = (const float*)d_in[0];
  const float* WB     = (const float*)d_in[1];
  const float* bB     = (const float*)d_in[2];
  const float* WC     = (const float*)d_in[3];
  const float* bC     = (const float*)d_in[4];
  const float* WD     = (const float*)d_in[5];
  const float* bD     = (const float*)d_in[6];
  const float* logA   = (const float*)d_in[7];
  const float* dparam = (const float*)d_in[8];

  float* out = (float*)d_out;
  float* ys  = out;                         // [L, D]
  float* hs  = out + (size_t)LSEQ * DCH;    // [L, D, N]

  // workspace layout (floats): Aprod | Hend | Hin | proj  (~12.6 MB)
  const size_t CS = (size_t)NCHUNK * DCH * NST;  // 1,048,576 floats per block
  float* ws    = (float*)d_ws;
  float* Aprod = ws;
  float* Hend  = Aprod + CS;
  float* Hin   = Hend + CS;
  float* proj  = Hin + CS;

  // 1) proj via WMMA: 128 tiles of 16 rows, one wave each
  proj_wmma_kernel<<<LSEQ / 16, 32, 0, stream>>>(xs, WD, bD, proj);

  // 2) chunk-local scans: grid (D/256, NCHUNK)
  dim3 gA(DCH / 256, NCHUNK);
  phaseA_kernel<<<gA, 256, 0, stream>>>(xs, proj, WB, bB, logA, dparam,
                                        Aprod, Hend);

  // 3) cross-chunk prefix per (d,n)
  phaseB_kernel<<<(DCH * NST) / 256, 256, 0, stream>>>(Aprod, Hend, Hin);

  // 4) final pass: write hs and ys
  phaseC_kernel<<<gA, 256, 0, stream>>>(xs, proj, WB, bB, WC, bC, logA, dparam,
                                        Hin, ys, hs);
}